// Model_32813550141884
// MI455X (gfx1250) — compile-verified
//
#include <hip/hip_runtime.h>
#include <hip/hip_bf16.h>

typedef float v2f __attribute__((ext_vector_type(2)));
typedef float v8f __attribute__((ext_vector_type(8)));

#define N_SEG 128
#define K_INIT 64
#define BZ 64
#define PPS (BZ * BZ)            // 4096 pixels per segment
#define EPS_F 1e-7f
#define CHUNKS 4
#define PIX_PER_CHUNK (PPS / CHUNKS)         // 1024
#define TILES_PER_CHUNK (PIX_PER_CHUNK / 16) // 64
#define WAVES 8
#define BLOCK (WAVES * 32)
#define LOG2E_F 1.44269504088896340736f

// One block = (segment, pixel-chunk). Exponents computed as a GEMM
// E[64k x 16p] = C[64 x 8] * G[8 x 16] via v_wmma_f32_16x16x4_f32.
// log2(e) is folded into C so the accumulator feeds v_exp_f32 (2^x) directly.
__global__ __launch_bounds__(BLOCK) void gsplat_seg_kernel(
    const float* __restrict__ u,   // [N_SEG, K_INIT, 2]
    const float* __restrict__ a,   // [N_SEG, K_INIT]
    const float* __restrict__ Bm,  // [N_SEG, K_INIT, 2, 2]
    float* __restrict__ out)       // [N_SEG, BZ, BZ]
{
    __shared__ float coef[K_INIT][8];  // per-component polynomial coeffs (K padded to 8)
    __shared__ float s_a[K_INIT];

    const int seg   = blockIdx.x >> 2;   // / CHUNKS
    const int chunk = blockIdx.x & 3;
    const int tid   = threadIdx.x;

    // ---- Stage 1: expand quadratic form into polynomial coefficients ----
    // exp(-0.5*||tril(B) d||^2) with d=(x-ux, y-uy):
    //   s00 = b00^2 + b10^2, s01 = b10*b11, s11 = b11^2
    //   log2-domain exponent: (c0 + c1 x + c2 y + c3 x^2 + c4 xy + c5 y^2),
    //   with (-0.5 * log2 e) folded into every coefficient.
    if (tid < K_INIT) {
        const int k = tid;
        const float ux = u[(seg * K_INIT + k) * 2 + 0];
        const float uy = u[(seg * K_INIT + k) * 2 + 1];
        const float* bp = Bm + (size_t)(seg * K_INIT + k) * 4;
        const float b00 = bp[0];
        // bp[1] (upper-right) is zeroed by tril
        const float b10 = bp[2];
        const float b11 = bp[3];
        const float s00 = b00 * b00 + b10 * b10;
        const float s01 = b10 * b11;
        const float s11 = b11 * b11;
        const float h = -0.5f * LOG2E_F;   // scale folded into all coefficients
        coef[k][0] = h * (s00 * ux * ux + 2.0f * s01 * ux * uy + s11 * uy * uy);
        coef[k][1] = -2.0f * h * (s00 * ux + s01 * uy);
        coef[k][2] = -2.0f * h * (s01 * ux + s11 * uy);
        coef[k][3] = h * s00;
        coef[k][4] = 2.0f * h * s01;
        coef[k][5] = h * s11;
        coef[k][6] = 0.0f;                  // K padding
        coef[k][7] = 0.0f;
        s_a[k] = a[seg * K_INIT + k];
    }
    __syncthreads();

    const int lane = tid & 31;
    const int wave = tid >> 5;
    const int hi   = lane >> 4;   // lane half: selects K {0,1} vs {2,3} per ISA layout
    const int l16  = lane & 15;

    // ---- A fragments (coefficient matrix), ISA 16x4 f32 A layout ----
    // VGPR0: K = 4g + 2*hi ; VGPR1: K = 4g + 1 + 2*hi ; M = l16 within tile.
    v2f  afrag[4][2];
    float aval[4][8];
    #pragma unroll
    for (int t = 0; t < 4; ++t) {
        const int row = t * 16 + l16;
        #pragma unroll
        for (int g = 0; g < 2; ++g) {
            afrag[t][g].x = coef[row][4 * g + 2 * hi];
            afrag[t][g].y = coef[row][4 * g + 2 * hi + 1];
        }
        // amplitudes matching C/D layout: VGPR r of tile t holds M = 16t + r + 8*hi
        #pragma unroll
        for (int r = 0; r < 8; ++r)
            aval[t][r] = s_a[t * 16 + r + 8 * hi];
    }

    const float invbz = 1.0f / (float)BZ;
    float* outp = out + (size_t)seg * PPS;

    for (int tile = wave; tile < TILES_PER_CHUNK; tile += WAVES) {
        const int p = chunk * PIX_PER_CHUNK + tile * 16 + l16;  // pixel index (N = l16)
        const float x = ((float)(p >> 6) + 0.5f) * invbz;       // meshgrid 'ij': x = ax[i]
        const float y = ((float)(p & 63) + 0.5f) * invbz;       //               y = ax[j]

        // ---- B fragments (pixel features), ISA 4x16 f32 B layout ----
        // g=0 rows {1, x | y, x^2} ; g=1 rows {xy, y^2 | 0, 0}
        v2f bg0, bg1;
        bg0.x = hi ? y       : 1.0f;
        bg0.y = hi ? x * x   : x;
        bg1.x = hi ? 0.0f    : x * y;
        bg1.y = hi ? 0.0f    : y * y;

        // ---- E = C x G via WMMA: 4 M-tiles x 2 K-groups ----
        v8f acc[4];
        #pragma unroll
        for (int t = 0; t < 4; ++t) {
            v8f c = {};
            c = __builtin_amdgcn_wmma_f32_16x16x4_f32(
                    false, afrag[t][0], false, bg0, (short)0, c, false, false);
            c = __builtin_amdgcn_wmma_f32_16x16x4_f32(
                    false, afrag[t][1], false, bg1, (short)0, c, false, false);
            acc[t] = c;
        }

        // ---- 2^E (log2e pre-folded), per-pixel normalization over k, clipped sum ----
        float e[4][8];
        float sum = 0.0f;
        #pragma unroll
        for (int t = 0; t < 4; ++t) {
            #pragma unroll
            for (int r = 0; r < 8; ++r) {
                const float v = __builtin_amdgcn_exp2f(acc[t][r]);  // v_exp_f32, no prescale
                e[t][r] = v;
                sum += v;
            }
        }
        // this lane holds half the M range; partner lane (xor 16) holds the rest
        const float ksum = fmaxf(sum + __shfl_xor(sum, 16, 32), EPS_F);
        const float rinv = 1.0f / ksum;

        float z = 0.0f;
        #pragma unroll
        for (int t = 0; t < 4; ++t) {
            #pragma unroll
            for (int r = 0; r < 8; ++r)
                z += fminf(aval[t][r] * e[t][r] * rinv, 1.0f);  // clip(.,0,1): all terms >= 0
        }
        z += __shfl_xor(z, 16, 32);

        if (hi == 0)
            outp[p] = z;
    }
}

extern "C" void kernel_launch(void* const* d_in, const int* in_sizes, int n_in,
                              void* d_out, int out_size, void* d_ws, size_t ws_size,
                              hipStream_t stream) {
    (void)in_sizes; (void)n_in; (void)out_size; (void)d_ws; (void)ws_size;
    const float* u  = (const float*)d_in[0];  // [128,64,2]
    const float* a  = (const float*)d_in[1];  // [128,64]
    const float* Bm = (const float*)d_in[2];  // [128,64,2,2]
    float* out = (float*)d_out;               // [128,64,64]

    dim3 grid(N_SEG * CHUNKS);
    dim3 block(BLOCK);
    gsplat_seg_kernel<<<grid, block, 0, stream>>>(u, a, Bm, out);
}